// FactorizedVectorQuantize_28535762714686
// MI455X (gfx1250) — compile-verified
//
#include <hip/hip_runtime.h>
#include <hip/hip_bf16.h>
#include <math.h>

// ---------------------------------------------------------------------------
// FactorizedVectorQuantize for MI455X (gfx1250, wave32)
//
// B=8, T=2048, D_in=1024, D_c=8, CB=8192.
// Memory-bound (~130MB HBM traffic vs 2.7 GFLOP fp32) -> stay fp32.
// - in-projection:  split-K (8 chunks) for memory-level parallelism,
//                   partials reduced by a tiny second kernel.
// - code search:    V_WMMA_F32_16X16X4_F32 x2 per 16x16 tile (K=8),
//                   per-lane running argmax + shfl_xor reduction.
// - out-projection: split over o-chunks (no reduction), coalesced stores.
// ---------------------------------------------------------------------------

#define Bsz   8
#define Tsz   2048
#define DIN   1024
#define DC    8
#define CB    8192
#define NTOK  (Bsz * Tsz)          // 16384
#define KC    8                    // split-K chunks for in-projection
#define KLEN  (DIN / KC)           // 128 i-values per chunk
#define OC    8                    // o-chunks for out-projection
#define OLEN  (DIN / OC)           // 128 o-values per chunk
#define ZESZ  (Bsz * DC * Tsz)     // 131072

typedef float v2f __attribute__((ext_vector_type(2)));
typedef float v8f __attribute__((ext_vector_type(8)));

// workspace layout (float offsets)
#define WS_WIN   0                         // [1024][8]  w_in TRANSPOSED (i-major)
#define WS_WOUT  (WS_WIN  + DIN * DC)      // [1024][8]  w_out (o-major)
#define WS_CBN   (WS_WOUT + DIN * DC)      // [8192][8]  normalized codebook
#define WS_PART  (WS_CBN  + CB * DC)       // [KC][131072] split-K partials (4MB)

// output layout (element offsets; return order: z_q_out, commit, cb_loss, indices, z_e)
#define OUT_ZQ   ((size_t)0)                       // 8*1024*2048
#define OUT_CL   (OUT_ZQ + (size_t)Bsz*DIN*Tsz)    // 8
#define OUT_BL   (OUT_CL + 8)                      // 8
#define OUT_IDX  (OUT_BL + 8)                      // 16384 (int32)
#define OUT_ZE   (OUT_IDX + NTOK)                  // 8*8*2048, layout [B][8][T]

// ---------------------------------------------------------------------------
// prep 1: w_inT[i][c] = in_v[c][i] * in_g[c] / ||in_v[c,:]||  (rows len 1024)
//         transposed store -> contiguous 32B rows for LDS b128 broadcasts.
//         Also zeroes the two loss outputs.
// ---------------------------------------------------------------------------
__global__ void k_prep_win(const float* __restrict__ in_v,
                           const float* __restrict__ in_g,
                           float* __restrict__ ws,
                           float* __restrict__ outf) {
    __shared__ float red[256];
    __shared__ float scale_s;
    float* w_inT = ws + WS_WIN;
    const int tid = threadIdx.x;

    if (tid < 16) outf[OUT_CL + tid] = 0.0f;   // commit & codebook losses = 0

    for (int c = 0; c < DC; ++c) {
        float s = 0.0f;
        #pragma unroll
        for (int k = 0; k < DIN / 256; ++k) {
            float v = in_v[c * DIN + k * 256 + tid];
            s += v * v;
        }
        red[tid] = s;
        __syncthreads();
        for (int off = 128; off > 0; off >>= 1) {
            if (tid < off) red[tid] += red[tid + off];
            __syncthreads();
        }
        if (tid == 0) scale_s = in_g[c] / sqrtf(red[0]);
        __syncthreads();
        float sc = scale_s;
        #pragma unroll
        for (int k = 0; k < DIN / 256; ++k) {
            int i = k * 256 + tid;
            w_inT[i * DC + c] = in_v[c * DIN + i] * sc;   // transposed
        }
        __syncthreads();
    }
}

// ---------------------------------------------------------------------------
// prep 2: w_out[o][c] = out_v[o][c] * out_g[o] / ||out_v[o,:]|| (rows len 8)
// ---------------------------------------------------------------------------
__global__ void k_prep_wout(const float* __restrict__ out_v,
                            const float* __restrict__ out_g,
                            float* __restrict__ ws) {
    int o = blockIdx.x * blockDim.x + threadIdx.x;
    if (o >= DIN) return;
    float v[DC];
    float s = 0.0f;
    #pragma unroll
    for (int c = 0; c < DC; ++c) { v[c] = out_v[o * DC + c]; s += v[c] * v[c]; }
    float sc = out_g[o] / sqrtf(s);
    float* w_out = ws + WS_WOUT;
    #pragma unroll
    for (int c = 0; c < DC; ++c) w_out[o * DC + c] = v[c] * sc;
}

// ---------------------------------------------------------------------------
// prep 3: cb_n = codebook / max(||row||, 1e-12)
// ---------------------------------------------------------------------------
__global__ void k_prep_cbn(const float* __restrict__ cb,
                           float* __restrict__ ws) {
    int r = blockIdx.x * blockDim.x + threadIdx.x;
    if (r >= CB) return;
    float v[DC];
    float s = 0.0f;
    #pragma unroll
    for (int c = 0; c < DC; ++c) { v[c] = cb[r * DC + c]; s += v[c] * v[c]; }
    float inv = 1.0f / fmaxf(sqrtf(s), 1e-12f);
    float* cbn = ws + WS_CBN;
    #pragma unroll
    for (int c = 0; c < DC; ++c) cbn[r * DC + c] = v[c] * inv;
}

// ---------------------------------------------------------------------------
// in-projection, split-K partial:
//   part[kc][b, c, t] = sum_{i in chunk kc} w_inT[i][c] * z[b, i, t]
// grid (T/256, B, KC): 4096 waves -> enough loads in flight for 23 TB/s.
// Weight slice (4KB) in LDS, read as two b128 broadcasts per i.
// ---------------------------------------------------------------------------
__global__ void k_in_partial(const float* __restrict__ z,
                             const float* __restrict__ ws,
                             float* __restrict__ part) {
    __shared__ float wl[KLEN * DC];          // 4 KB weight slice
    const int kc = blockIdx.z;
    const int i0 = kc * KLEN;
    const float* w_inT = ws + WS_WIN + (size_t)i0 * DC;
    for (int i = threadIdx.x; i < KLEN * DC; i += blockDim.x) wl[i] = w_inT[i];
    __syncthreads();

    const int b = blockIdx.y;
    const int t = blockIdx.x * blockDim.x + threadIdx.x;
    const float* zp = z + (size_t)b * DIN * Tsz + (size_t)i0 * Tsz + t;

    float acc[DC];
    #pragma unroll
    for (int c = 0; c < DC; ++c) acc[c] = 0.0f;

    const float4* wv = (const float4*)wl;
    #pragma unroll 8
    for (int ii = 0; ii < KLEN; ++ii) {
        float zv = zp[(size_t)ii * Tsz];
        float4 w0 = wv[ii * 2];
        float4 w1 = wv[ii * 2 + 1];
        acc[0] += w0.x * zv;  acc[1] += w0.y * zv;
        acc[2] += w0.z * zv;  acc[3] += w0.w * zv;
        acc[4] += w1.x * zv;  acc[5] += w1.y * zv;
        acc[6] += w1.z * zv;  acc[7] += w1.w * zv;
    }

    float* pp = part + (size_t)kc * ZESZ + (size_t)b * DC * Tsz + t;
    #pragma unroll
    for (int c = 0; c < DC; ++c) pp[(size_t)c * Tsz] = acc[c];
}

// ---------------------------------------------------------------------------
// reduce split-K partials + bias -> z_e output region [B][8][T]
// ---------------------------------------------------------------------------
__global__ void k_ze_reduce(const float* __restrict__ part,
                            const float* __restrict__ in_b,
                            float* __restrict__ outf) {
    const int idx = blockIdx.x * blockDim.x + threadIdx.x;   // [0, 131072)
    const int c = (idx >> 11) & 7;                           // layout b*8T + c*T + t
    float s = in_b[c];
    #pragma unroll
    for (int kc = 0; kc < KC; ++kc) s += part[(size_t)kc * ZESZ + idx];
    outf[OUT_ZE + idx] = s;
}

// ---------------------------------------------------------------------------
// codebook search: indices[tok] = argmax_n  z_e[tok] . cb_n[n]
// (== argmin normalized-L2; z_e normalization is a positive row scale and
//  cannot change the argmax, so raw z_e is the A operand.)
//
// One wave per 16-token tile; A held in 4 VGPRs for the 8192-code sweep.
// Per 16-code chunk: two V_WMMA_F32_16X16X4_F32 (K=0..3, then K=4..7).
// F32 16x16x4 operand layout (wave32):
//   A/B: VGPR0 = K0 (lanes 0-15) / K2 (lanes 16-31); VGPR1 = K1 / K3
//   D:   VGPRr = M=r (lanes 0-15) / M=r+8 (lanes 16-31), N = lane%16
// ---------------------------------------------------------------------------
__global__ void k_dist_argmax(const float* __restrict__ ze,   // [B][8][T]
                              const float* __restrict__ cbn,  // [8192][8]
                              int* __restrict__ idx_out) {
    const int lane  = threadIdx.x & 31;
    const int wave  = threadIdx.x >> 5;
    const int tile  = blockIdx.x * (blockDim.x >> 5) + wave;
    const int m0    = tile * 16;
    const int row   = lane & 15;
    const int khalf = lane >> 4;     // 0 -> K0,K1 ; 1 -> K2,K3

    // A operand straight from the z_e output region (strided, once per sweep)
    const int tok = m0 + row;
    const int b   = tok >> 11;           // /T
    const int t   = tok & (Tsz - 1);
    const float* zp = ze + (size_t)b * DC * Tsz + t;
    const int kb = khalf * 2;
    v2f a0 = { zp[(size_t)(kb + 0) * Tsz], zp[(size_t)(kb + 1) * Tsz] };
    v2f a1 = { zp[(size_t)(kb + 4) * Tsz], zp[(size_t)(kb + 5) * Tsz] };

    float bestV[8];
    int   bestI[8];
    #pragma unroll
    for (int r = 0; r < 8; ++r) { bestV[r] = -INFINITY; bestI[r] = 0; }

    #pragma unroll 4
    for (int n0 = 0; n0 < CB; n0 += 16) {
        const float* cp = cbn + (size_t)(n0 + row) * DC + kb;
        v2f b0 = { cp[0], cp[1] };
        v2f b1 = { cp[4], cp[5] };
        v8f d = {0.f, 0.f, 0.f, 0.f, 0.f, 0.f, 0.f, 0.f};
        d = __builtin_amdgcn_wmma_f32_16x16x4_f32(false, a0, false, b0,
                                                  (short)0, d, false, false);
        d = __builtin_amdgcn_wmma_f32_16x16x4_f32(false, a1, false, b1,
                                                  (short)0, d, false, false);
        const int n = n0 + row;      // this lane's N index
        #pragma unroll
        for (int r = 0; r < 8; ++r) {
            float v = d[r];
            if (v > bestV[r]) { bestV[r] = v; bestI[r] = n; }  // strict >: lowest n wins
        }
    }

    // argmax across the 16 lanes of each half (smaller index wins ties)
    #pragma unroll
    for (int r = 0; r < 8; ++r) {
        float v = bestV[r];
        int   i = bestI[r];
        #pragma unroll
        for (int m = 1; m < 16; m <<= 1) {
            float ov = __shfl_xor(v, m, 32);
            int   oi = __shfl_xor(i, m, 32);
            if (ov > v || (ov == v && oi < i)) { v = ov; i = oi; }
        }
        if (row == 0) idx_out[m0 + r + khalf * 8] = i;   // M = r (+8 for hi half)
    }
}

// ---------------------------------------------------------------------------
// out-projection: z_q_out[b,o,t] = sum_c w_out[o,c]*codebook[idx[b,t],c] + out_b[o]
// grid (T/256, B, OC): each block owns a 128-wide o-slice -> 4096 waves of
// coalesced stores. Weight slice + bias in LDS (b128 broadcasts).
// ---------------------------------------------------------------------------
__global__ void k_out_project(const float* __restrict__ cb,
                              const float* __restrict__ out_b,
                              const float* __restrict__ ws,
                              const int* __restrict__ idx,
                              float* __restrict__ outf) {
    __shared__ float wl[OLEN * DC];   // 4 KB
    __shared__ float bl[OLEN];        // 512 B
    const int o0 = blockIdx.z * OLEN;
    const float* w_out = ws + WS_WOUT + (size_t)o0 * DC;
    for (int i = threadIdx.x; i < OLEN * DC; i += blockDim.x) wl[i] = w_out[i];
    for (int i = threadIdx.x; i < OLEN; i += blockDim.x) bl[i] = out_b[o0 + i];
    __syncthreads();

    const int b = blockIdx.y;
    const int t = blockIdx.x * blockDim.x + threadIdx.x;
    const int code = idx[(size_t)b * Tsz + t];

    const float4* qp = (const float4*)(cb + (size_t)code * DC);
    float4 q0 = qp[0], q1 = qp[1];

    float* op = outf + OUT_ZQ + (size_t)b * DIN * Tsz + (size_t)o0 * Tsz + t;
    const float4* wv = (const float4*)wl;
    #pragma unroll 4
    for (int oo = 0; oo < OLEN; ++oo) {
        float4 w0 = wv[oo * 2];
        float4 w1 = wv[oo * 2 + 1];
        float acc = bl[oo];
        acc += w0.x * q0.x + w0.y * q0.y + w0.z * q0.z + w0.w * q0.w;
        acc += w1.x * q1.x + w1.y * q1.y + w1.z * q1.z + w1.w * q1.w;
        op[(size_t)oo * Tsz] = acc;
    }
}

// ---------------------------------------------------------------------------
extern "C" void kernel_launch(void* const* d_in, const int* in_sizes, int n_in,
                              void* d_out, int out_size, void* d_ws, size_t ws_size,
                              hipStream_t stream) {
    const float* z        = (const float*)d_in[0];
    const float* in_v     = (const float*)d_in[1];
    const float* in_g     = (const float*)d_in[2];
    const float* in_b     = (const float*)d_in[3];
    const float* out_v    = (const float*)d_in[4];
    const float* out_g    = (const float*)d_in[5];
    const float* out_b    = (const float*)d_in[6];
    const float* codebook = (const float*)d_in[7];

    float* outf = (float*)d_out;
    float* ws   = (float*)d_ws;
    int*   idxp = (int*)(outf + OUT_IDX);   // int32 argmax indices segment

    // weight prep + codebook normalization (tiny)
    k_prep_win <<<1, 256, 0, stream>>>(in_v, in_g, ws, outf);
    k_prep_wout<<<DIN / 256, 256, 0, stream>>>(out_v, out_g, ws);
    k_prep_cbn <<<CB  / 256, 256, 0, stream>>>(codebook, ws);

    // z_e = w_in @ z + in_b : split-K partials (64MB read, 4096 waves)
    k_in_partial<<<dim3(Tsz / 256, Bsz, KC), 256, 0, stream>>>(z, ws, ws + WS_PART);
    k_ze_reduce <<<ZESZ / 256, 256, 0, stream>>>(ws + WS_PART, in_b, outf);

    // nearest-code search via fp32 WMMA (1024 waves x 512 x 2 wmma)
    k_dist_argmax<<<(NTOK / 16) / 8, 256, 0, stream>>>(outf + OUT_ZE, ws + WS_CBN, idxp);

    // z_q_out = w_out @ codebook[idx] + out_b (64MB store, 4096 waves)
    k_out_project<<<dim3(Tsz / 256, Bsz, OC), 256, 0, stream>>>(codebook, out_b, ws,
                                                                idxp, outf);
}